// SinkhornKnopp_75411035783864
// MI455X (gfx1250) — compile-verified
//
#include <hip/hip_runtime.h>
#include <math.h>

// Sinkhorn-Knopp, restructured as diagonal-rescale form:
//   out[b,k] = E[b,k] * u3[k] / s3[b],  E = exp(logits/0.05)
// 4 streaming passes over logits (memory-optimal: ~2.5GB @ 23.3TB/s ~ 107us).
// Row dot-products s[b] = sum_k E[b,k]*u[k] computed with V_WMMA_F32_16X16X4_F32.

typedef __attribute__((ext_vector_type(2))) float v2f;
typedef __attribute__((ext_vector_type(8))) float v8f;

#define SK_K 512
#define SK_B 262144
#define SK_INV_EPS 20.0f           // 1/0.05
#define TILE_ROWS 16
#define LDS_STRIDE 516             // 512 + 4 pad -> conflict-free strided reads
#define WAVES_PER_WG 4
#define WG_COUNT 1024
#define TILES_PER_WAVE 4           // 16384 tiles / (1024 WGs * 4 waves)

__global__ void sk_zero(float* t) {
  const int i = threadIdx.x;       // 512 threads
  t[i] = 0.f; t[i + 512] = 0.f; t[i + 1024] = 0.f;
}

// Pass 1: t1[k] = sum_b exp(logits[b,k]/eps)   (initial global sum cancels)
__global__ void sk_pass1(const float* __restrict__ logits, float* __restrict__ t1) {
  const int k = threadIdx.x;                   // 512 threads: thread <-> column
  const int rows = SK_B / 2048;                // grid = 2048 blocks -> 128 rows each
  const float* p = logits + (size_t)blockIdx.x * rows * SK_K + k;
  float acc = 0.f;
  for (int r = 0; r < rows; ++r)
    acc += __expf(p[(size_t)r * SK_K] * SK_INV_EPS);
  atomicAdd(&t1[k], acc);
}

// Fused pass. MODE 0: compute v[b] locally, accumulate t_out[k] += E*v.
//             MODE 1: compute 1/s[b] locally, write out = E*u/s.
template <int MODE>
__global__ void sk_fused(const float* __restrict__ logits,
                         const float* __restrict__ t_in,
                         float* __restrict__ t_out,
                         float* __restrict__ outp) {
  extern __shared__ float lds[];
  float* u = lds;                                        // 512 floats, shared by WG
  const int tid  = threadIdx.x;
  const int lane = tid & 31;
  const int wave = tid >> 5;
  float* Et = lds + SK_K + wave * (TILE_ROWS * LDS_STRIDE);  // per-wave tile

  // u[k] = 1 / (K * t_in[k])
  for (int k = tid; k < SK_K; k += WAVES_PER_WG * 32)
    u[k] = 1.0f / (512.0f * t_in[k]);
  __syncthreads();

  float treg[16];
  if (MODE == 0) {
#pragma unroll
    for (int i = 0; i < 16; ++i) treg[i] = 0.f;
  }

  const int rowA = lane & 15;                 // A-matrix row for this lane
  const int off  = (lane >> 4) << 1;          // K sub-chunk: 0 (lanes 0-15) / 2 (16-31)
  const int gwave = blockIdx.x * WAVES_PER_WG + wave;

  for (int j = 0; j < TILES_PER_WAVE; ++j) {
    const size_t row0 = ((size_t)gwave * TILES_PER_WAVE + j) * TILE_ROWS;

    // ---- stage E = exp(logits/eps) tile into LDS (coalesced b128 loads) ----
    for (int r = 0; r < TILE_ROWS; ++r) {
      const float* src = logits + (row0 + r) * (size_t)SK_K;
      float* dst = Et + r * LDS_STRIDE;
#pragma unroll
      for (int it = 0; it < 4; ++it) {
        const int k = it * 128 + lane * 4;
        float4 x = *reinterpret_cast<const float4*>(src + k);
        float4 e;
        e.x = __expf(x.x * SK_INV_EPS);
        e.y = __expf(x.y * SK_INV_EPS);
        e.z = __expf(x.z * SK_INV_EPS);
        e.w = __expf(x.w * SK_INV_EPS);
        *reinterpret_cast<float4*>(dst + k) = e;
      }
    }
    __syncthreads();

    // ---- s[m] = sum_k E[m,k]*u[k] via V_WMMA_F32_16X16X4_F32 ----
    // A: 16x4 f32 (lane<16: M=lane, K=k0+0/1 ; lane>=16: M=lane-16, K=k0+2/3)
    // B: 4x16 f32, every column = u segment -> every C column holds s[m].
    v8f c0 = {}; v8f c1 = {};
    const float* arow = Et + rowA * LDS_STRIDE;
#pragma unroll 4
    for (int k0 = 0; k0 < SK_K; k0 += 8) {
      v2f a0 = *reinterpret_cast<const v2f*>(arow + k0 + off);
      v2f b0 = *reinterpret_cast<const v2f*>(u + k0 + off);
      c0 = __builtin_amdgcn_wmma_f32_16x16x4_f32(false, a0, false, b0,
                                                 (short)0, c0, false, false);
      v2f a1 = *reinterpret_cast<const v2f*>(arow + k0 + 4 + off);
      v2f b1 = *reinterpret_cast<const v2f*>(u + k0 + 4 + off);
      c1 = __builtin_amdgcn_wmma_f32_16x16x4_f32(false, a1, false, b1,
                                                 (short)0, c1, false, false);
    }
    v8f c = c0 + c1;

    // ---- broadcast per-row scale (C layout: VGPR j = rows j / j+8) ----
    float vv[16];
#pragma unroll
    for (int r = 0; r < 16; ++r) {
      float sr = __shfl(c[r & 7], (r < 8) ? 0 : 16, 32);
      vv[r] = (MODE == 0) ? 1.0f / ((float)SK_B * sr)   // v[b] = 1/(B*s)
                          : 1.0f / sr;                  // final: 1/s
    }

    if (MODE == 0) {
      // t_out[k] += sum_r E[r,k] * v[r]   (lane owns k = lane+32i)
#pragma unroll
      for (int r = 0; r < 16; ++r) {
        const float* er = Et + r * LDS_STRIDE;
        const float vr = vv[r];
#pragma unroll
        for (int i = 0; i < 16; ++i)
          treg[i] += er[lane + 32 * i] * vr;
      }
    } else {
      // out[b,k] = E[b,k] * u[k] / s[b]   (coalesced b128 stores)
      float* orow = outp + row0 * (size_t)SK_K;
#pragma unroll
      for (int r = 0; r < 16; ++r) {
        const float* er = Et + r * LDS_STRIDE;
        const float is = vv[r];
        for (int it = 0; it < 4; ++it) {
          const int k = it * 128 + lane * 4;
          float4 w;
          w.x = er[k + 0] * u[k + 0] * is;
          w.y = er[k + 1] * u[k + 1] * is;
          w.z = er[k + 2] * u[k + 2] * is;
          w.w = er[k + 3] * u[k + 3] * is;
          *reinterpret_cast<float4*>(orow + (size_t)r * SK_K + k) = w;
        }
      }
    }
    __syncthreads();
  }

  if (MODE == 0) {
#pragma unroll
    for (int i = 0; i < 16; ++i)
      atomicAdd(&t_out[lane + 32 * i], treg[i]);   // 2KB array, L2-resident
  }
}

extern "C" void kernel_launch(void* const* d_in, const int* in_sizes, int n_in,
                              void* d_out, int out_size, void* d_ws, size_t ws_size,
                              hipStream_t stream) {
  (void)in_sizes; (void)n_in; (void)out_size; (void)ws_size;
  const float* logits = (const float*)d_in[0];
  float* out = (float*)d_out;
  float* t1 = (float*)d_ws;          // 3 x 512 floats of scratch
  float* t2 = t1 + 512;
  float* t3 = t1 + 1024;

  sk_zero<<<1, 512, 0, stream>>>(t1);
  sk_pass1<<<2048, 512, 0, stream>>>(logits, t1);

  const size_t ldsBytes =
      (SK_K + WAVES_PER_WG * TILE_ROWS * LDS_STRIDE) * sizeof(float);  // ~131 KB (<320KB/WG)
  sk_fused<0><<<WG_COUNT, WAVES_PER_WG * 32, ldsBytes, stream>>>(logits, t1, t2, nullptr);
  sk_fused<0><<<WG_COUNT, WAVES_PER_WG * 32, ldsBytes, stream>>>(logits, t2, t3, nullptr);
  sk_fused<1><<<WG_COUNT, WAVES_PER_WG * 32, ldsBytes, stream>>>(logits, t3, nullptr, out);
}